// ModelNew_17514876633436
// MI455X (gfx1250) — compile-verified
//
#include <hip/hip_runtime.h>

// ---------------------------------------------------------------------------
// Bidirectional 3-layer GRU for MI455X (gfx1250), fp32 via V_WMMA_F32_16X16X4_F32
//
// L=3, H=256, IN=512, S=512, B=64.
// Phase 1 per layer: gi[d,s,b,g] = x·Wih^T + b_ih  — WMMA GEMM, 2x4 register
//   blocking per wave (A reused 4x, B reused 2x).
// Phase 2 per layer: sequential scan, gh = h·Whh^T + b_hh via WMMA with the
//   per-wave Whh slice resident in LDS in B-fragment order (prepacked once,
//   reused for all 512 steps). h exchanged via global hbuf + per-group atomic
//   barrier (4 n-slice blocks per (dir,bslice) group).
// ---------------------------------------------------------------------------

typedef __attribute__((ext_vector_type(2))) float v2f;
typedef __attribute__((ext_vector_type(4))) float v4f;
typedef __attribute__((ext_vector_type(8))) float v8f;

#define SEQ 512
#define BAT 64
#define HID 256
#define GCOLS 768            // 3*HID
#define INW 512              // input width (== 2*HID for every layer)
#define ROWS (SEQ * BAT)     // 32768 GEMM rows for gi

__device__ __forceinline__ float sig_f(float x) {
    return 1.0f / (1.0f + __expf(-x));
}
__device__ __forceinline__ float tanh_f(float x) {
    return 2.0f / (1.0f + __expf(-2.0f * x)) - 1.0f;
}

// ---------------------------------------------------------------------------
// Kernel 1: gi = X(32768x512) * Wih^T(512x768) + b_ih, both directions.
// Each wave computes a 32(M) x 64(N) block = 2x4 WMMA tiles, K streamed in 4s.
// Jobs per dir: (32768/32) * (768/64) = 1024*12 = 12288; 8 waves/block
// -> grid = (1536, 2), block = 256.
// ---------------------------------------------------------------------------
__global__ __launch_bounds__(256) void gi_gemm(
    const float* __restrict__ X,     // [ROWS][INW]
    const float* __restrict__ Wih,   // [2][GCOLS][INW]   (layer base)
    const float* __restrict__ bih,   // [2][GCOLS]        (layer base)
    float* __restrict__ gi)          // [2][ROWS][GCOLS]
{
    const int lane = threadIdx.x & 31;
    const int wave = threadIdx.x >> 5;
    const int d    = blockIdx.y;
    const int t    = blockIdx.x * 8 + wave;     // job id, 12288 per dir
    const int rowT = t / 12;
    const int colT = t % 12;
    const int r0   = rowT << 5;                 // 32 rows
    const int g0   = colT << 6;                 // 64 cols
    const int l15  = lane & 15;
    const int kof  = (lane >> 4) << 1;          // K sub-offset per WMMA layout
    const int mof  = (lane >> 4) << 3;          // C/D: lanes 16-31 hold M+8

    const float* __restrict__ a0p = X + (size_t)(r0 + l15) * INW + kof;
    const float* __restrict__ a1p = a0p + (size_t)16 * INW;
    const float* __restrict__ wbase =
        Wih + (size_t)d * GCOLS * INW + (size_t)(g0 + l15) * INW + kof;

    v8f acc[2][4];
#pragma unroll
    for (int mt = 0; mt < 2; ++mt)
#pragma unroll
        for (int nt = 0; nt < 4; ++nt)
            acc[mt][nt] = (v8f){};

#pragma unroll 4
    for (int k = 0; k < INW; k += 4) {
        v2f a0 = *(const v2f*)(a0p + k);
        v2f a1 = *(const v2f*)(a1p + k);
        v2f b0 = *(const v2f*)(wbase + k);
        v2f b1 = *(const v2f*)(wbase + 16 * INW + k);
        v2f b2 = *(const v2f*)(wbase + 32 * INW + k);
        v2f b3 = *(const v2f*)(wbase + 48 * INW + k);
        acc[0][0] = __builtin_amdgcn_wmma_f32_16x16x4_f32(false, a0, false, b0, (short)0, acc[0][0], false, false);
        acc[0][1] = __builtin_amdgcn_wmma_f32_16x16x4_f32(false, a0, false, b1, (short)0, acc[0][1], false, false);
        acc[0][2] = __builtin_amdgcn_wmma_f32_16x16x4_f32(false, a0, false, b2, (short)0, acc[0][2], false, false);
        acc[0][3] = __builtin_amdgcn_wmma_f32_16x16x4_f32(false, a0, false, b3, (short)0, acc[0][3], false, false);
        acc[1][0] = __builtin_amdgcn_wmma_f32_16x16x4_f32(false, a1, false, b0, (short)0, acc[1][0], false, false);
        acc[1][1] = __builtin_amdgcn_wmma_f32_16x16x4_f32(false, a1, false, b1, (short)0, acc[1][1], false, false);
        acc[1][2] = __builtin_amdgcn_wmma_f32_16x16x4_f32(false, a1, false, b2, (short)0, acc[1][2], false, false);
        acc[1][3] = __builtin_amdgcn_wmma_f32_16x16x4_f32(false, a1, false, b3, (short)0, acc[1][3], false, false);
    }

#pragma unroll
    for (int mt = 0; mt < 2; ++mt) {
#pragma unroll
        for (int nt = 0; nt < 4; ++nt) {
            const float bias = bih[d * GCOLS + g0 + nt * 16 + l15];
            float* __restrict__ op = gi
                + ((size_t)d * ROWS + r0 + mt * 16 + mof) * GCOLS
                + g0 + nt * 16 + l15;
#pragma unroll
            for (int v = 0; v < 8; ++v)
                op[(size_t)v * GCOLS] = acc[mt][nt][v] + bias;
        }
    }
}

// ---------------------------------------------------------------------------
// Kernel 2: recurrent scan for one layer, both directions.
// grid = (4 nslice, 4 bslice, 2 dir), block = 128 (4 waves).
// Wave w owns hidden cols [ns*64 + w*16, +16): three WMMA gate tiles (r,z,n).
// Whh slice lives in LDS in B-fragment order: per wave 3 gates x 64 ksteps x
// 64 floats (lane*2 consecutive -> conflict-free ds_load_b64).
// Static LDS: 4*3*64*64*4 = 196608 B weights + 16384 B h stage = 212992 B.
// ---------------------------------------------------------------------------
#define NSL 4        // n-slice blocks per (dir,bslice) barrier group
#define WFRAG (3 * 64 * 64)   // floats of fragment storage per wave

__global__ __launch_bounds__(128) void gru_scan(
    const float* __restrict__ gi,    // [2][ROWS][GCOLS]
    const float* __restrict__ Whh,   // [2][GCOLS][HID]  (layer base)
    const float* __restrict__ bhh,   // [2][GCOLS]       (layer base)
    const float* __restrict__ h0,    // [2][BAT][HID]    (layer base)
    float* __restrict__ y,           // [SEQ][BAT][2*HID]
    float* __restrict__ hbuf,        // [2][BAT][HID]
    unsigned* __restrict__ ctr,      // [8]
    float* __restrict__ hout)        // [2][BAT][HID]    (layer base of d_out)
{
    __shared__ float wlds[4 * WFRAG];   // 196608 B: per-wave Whh B-fragments
    __shared__ float hsl[16 * HID];     // 16384 B: staged h rows (batch slice)

    const int tid  = threadIdx.x;
    const int lane = tid & 31;
    const int wave = tid >> 5;         // 0..3
    const int ns   = blockIdx.x;       // 0..3
    const int bs   = blockIdx.y;       // 0..3
    const int d    = blockIdx.z;       // 0..1
    const int b0   = bs << 4;
    const int c0   = (ns << 6) + (wave << 4);
    const int l15  = lane & 15;
    const int kof  = (lane >> 4) << 1;
    const int mof  = (lane >> 4) << 3;
    const unsigned grp = (unsigned)(d * 4 + bs);

    const float* __restrict__ Wd = Whh + (size_t)d * GCOLS * HID;
    const float* __restrict__ bd = bhh + d * GCOLS;
    const float bias_r = bd[c0 + l15];
    const float bias_z = bd[HID + c0 + l15];
    const float bias_n = bd[2 * HID + c0 + l15];

    float*       hrow  = hbuf + d * BAT * HID + b0 * HID;  // [16][HID] region
    const float* h0row = h0   + d * BAT * HID + b0 * HID;

    // ---- prepack this wave's Whh slice into LDS in B-fragment order ----
    float* __restrict__ myw = wlds + wave * WFRAG;
#pragma unroll
    for (int g = 0; g < 3; ++g) {
        const float* __restrict__ ws =
            Wd + (size_t)(g * HID + c0 + l15) * HID + kof;
        float* __restrict__ wd2 = myw + g * (64 * 64) + lane * 2;
#pragma unroll 8
        for (int kk = 0; kk < 64; ++kk) {
            *(v2f*)(wd2 + kk * 64) = *(const v2f*)(ws + kk * 4);
        }
    }
    // (each wave reads only its own region; no cross-wave sync needed)

    for (int step = 0; step < SEQ; ++step) {
        const int s = d ? (SEQ - 1 - step) : step;

        // ---- stage h (16 rows x 256 cols, contiguous 16 KB) into LDS ----
        {
            const v4f* __restrict__ src =
                (const v4f*)(step == 0 ? h0row : (const float*)hrow);
            v4f* __restrict__ dst = (v4f*)hsl;
#pragma unroll
            for (int i = 0; i < 8; ++i)
                dst[tid + i * 128] = src[tid + i * 128];
        }
        __syncthreads();

        // ---- gh = h * Whh^T + b_hh via f32 WMMA, 3 gate tiles per wave ----
        v8f ar, az, an;
#pragma unroll
        for (int v = 0; v < 8; ++v) { ar[v] = bias_r; az[v] = bias_z; an[v] = bias_n; }

#pragma unroll 4
        for (int kk = 0; kk < 64; ++kk) {
            v2f a  = *(const v2f*)&hsl[l15 * HID + kk * 4 + kof];
            v2f br = *(const v2f*)&myw[(0 * 64 + kk) * 64 + lane * 2];
            v2f bz = *(const v2f*)&myw[(1 * 64 + kk) * 64 + lane * 2];
            v2f bn = *(const v2f*)&myw[(2 * 64 + kk) * 64 + lane * 2];
            ar = __builtin_amdgcn_wmma_f32_16x16x4_f32(false, a, false, br, (short)0, ar, false, false);
            az = __builtin_amdgcn_wmma_f32_16x16x4_f32(false, a, false, bz, (short)0, az, false, false);
            an = __builtin_amdgcn_wmma_f32_16x16x4_f32(false, a, false, bn, (short)0, an, false, false);
        }

        // ---- gates + state update for this wave's 16x16 h_new tile ----
        const size_t girow = ((size_t)d * ROWS + (size_t)s * BAT + b0 + mof) * GCOLS;
        const int col = c0 + l15;
#pragma unroll
        for (int v = 0; v < 8; ++v) {
            const int m = v + mof;
            const size_t gb = girow + (size_t)v * GCOLS;
            const float g_r = gi[gb + col];
            const float g_z = gi[gb + HID + col];
            const float g_n = gi[gb + 2 * HID + col];
            const float r = sig_f(g_r + ar[v]);
            const float z = sig_f(g_z + az[v]);
            const float n = tanh_f(g_n + r * an[v]);   // b_hh_n is inside r*(...)
            const float hold = hsl[m * HID + col];
            const float hnew = (1.0f - z) * n + z * hold;
            hrow[m * HID + col] = hnew;
            y[((size_t)s * BAT + b0 + m) * (2 * HID) + d * HID + col] = hnew;
            if (step == SEQ - 1)
                hout[d * BAT * HID + (b0 + m) * HID + col] = hnew;
        }

        // ---- exchange h among the 4 n-slice blocks of this (dir,bslice) ----
        if (step != SEQ - 1) {
            __threadfence();
            __syncthreads();
            if (tid == 0) {
                __hip_atomic_fetch_add(&ctr[grp], 1u, __ATOMIC_RELEASE,
                                       __HIP_MEMORY_SCOPE_AGENT);
                const unsigned target = (unsigned)NSL * (unsigned)(step + 1);
                while (__hip_atomic_load(&ctr[grp], __ATOMIC_ACQUIRE,
                                         __HIP_MEMORY_SCOPE_AGENT) < target) {
                    __builtin_amdgcn_s_sleep(1);
                }
            }
            __syncthreads();
        }
    }
}

// ---------------------------------------------------------------------------
// Host launcher. Workspace layout (all fp32 unless noted):
//   yA  [SEQ][BAT][512]   64 MB   ping activation buffer
//   yB  [SEQ][BAT][512]   64 MB   pong activation buffer
//   gi  [2][ROWS][768]   192 MB   per-layer input projections
//   hbuf[2][BAT][256]    128 KB   recurrent state exchange
//   ctr [8] u32                    per-group step barrier (memset each layer)
// ---------------------------------------------------------------------------
extern "C" void kernel_launch(void* const* d_in, const int* in_sizes, int n_in,
                              void* d_out, int out_size, void* d_ws, size_t ws_size,
                              hipStream_t stream) {
    (void)in_sizes; (void)n_in; (void)out_size; (void)ws_size;

    const float* x   = (const float*)d_in[0];  // [SEQ][BAT][INW]
    const float* h0  = (const float*)d_in[1];  // [2L][BAT][HID]
    const float* Wih = (const float*)d_in[2];  // [L][2][GCOLS][INW]
    const float* bih = (const float*)d_in[3];  // [L][2][GCOLS]
    const float* Whh = (const float*)d_in[4];  // [L][2][GCOLS][HID]
    const float* bhh = (const float*)d_in[5];  // [L][2][GCOLS]
    float* out = (float*)d_out;                // [2L][BAT][HID]

    float* yA   = (float*)d_ws;
    float* yB   = yA + (size_t)SEQ * BAT * 2 * HID;
    float* gi   = yB + (size_t)SEQ * BAT * 2 * HID;
    float* hbuf = gi + (size_t)2 * ROWS * GCOLS;
    unsigned* ctr = (unsigned*)(hbuf + (size_t)2 * BAT * HID);

    const dim3 gGemm(1536, 2, 1);      // 12288 wave-jobs/dir / 8 waves per block
    const dim3 gScan(4, 4, 2);         // nslice, bslice, dir

    const float* in = x;
    float* outs[3] = { yA, yB, yA };
    for (int l = 0; l < 3; ++l) {
        gi_gemm<<<gGemm, 256, 0, stream>>>(
            in,
            Wih + (size_t)l * 2 * GCOLS * INW,
            bih + (size_t)l * 2 * GCOLS,
            gi);
        hipMemsetAsync(ctr, 0, 8 * sizeof(unsigned), stream);
        gru_scan<<<gScan, 128, 0, stream>>>(
            gi,
            Whh + (size_t)l * 2 * GCOLS * HID,
            bhh + (size_t)l * 2 * GCOLS,
            h0  + (size_t)2 * l * BAT * HID,
            outs[l], hbuf, ctr,
            out + (size_t)2 * l * BAT * HID);
        in = outs[l];
    }
}